// SplitPool_45071386804902
// MI455X (gfx1250) — compile-verified
//
#include <hip/hip_runtime.h>

typedef float v2f __attribute__((ext_vector_type(2)));
typedef float v8f __attribute__((ext_vector_type(8)));

#define BATCH   8
#define LEN     32768
#define DIM     256
#define PMAX    400
#define PTILES  25          // 400 / 16
#define SPLIT   4           // row-splits per peak-tile (parallelism)
#define OUT_N   (BATCH * PMAX * DIM)

// ---------------------------------------------------------------------------
// Kernel 0: exclusive prefix sum of chunk_size -> chunkStart[totalChunks+1],
// plus per-sample first-chunk index ss[0..BATCH] (cumsum of n_peaks+1).
// ---------------------------------------------------------------------------
__global__ __launch_bounds__(256) void prefix_kernel(
    const int* __restrict__ cs, const int* __restrict__ np,
    int* __restrict__ chunkStart, int* __restrict__ ss, int totalChunks) {
  __shared__ int part[256];
  const int t = threadIdx.x;
  const int K = (totalChunks + 255) >> 8;
  const int b = t * K;
  const int e = (b + K < totalChunks) ? (b + K) : totalChunks;
  int sum = 0;
  for (int i = b; i < e; ++i) sum += cs[i];
  const int mine = sum;
  part[t] = sum;
  __syncthreads();
  // Hillis-Steele inclusive scan over 256 partials
  for (int off = 1; off < 256; off <<= 1) {
    int v = (t >= off) ? part[t - off] : 0;
    __syncthreads();
    part[t] += v;
    __syncthreads();
  }
  int run = part[t] - mine;  // exclusive prefix for this thread's range
  for (int i = b; i < e; ++i) { chunkStart[i] = run; run += cs[i]; }
  if (b < totalChunks && e == totalChunks) chunkStart[totalChunks] = run;
  if (t == 0) {
    int acc = 0;
    ss[0] = 0;
    for (int i = 0; i < BATCH; ++i) { acc += np[i] + 1; ss[i + 1] = acc; }
  }
}

// ---------------------------------------------------------------------------
// Kernel 1 (fallback path only): zero the output.
// ---------------------------------------------------------------------------
__global__ __launch_bounds__(256) void zero_kernel(float* __restrict__ out) {
  out[(size_t)blockIdx.x * 256 + threadIdx.x] = 0.0f;
}

// ---------------------------------------------------------------------------
// Kernel 2: segmented sum via V_WMMA_F32_16X16X4_F32.
//   D(16 chunks x 16 cols) += A(16 chunks x 4 rows, one-hot) * B(4 rows x 16 cols)
// Block = 256 thr (8 waves). Wave w owns columns [32w, 32w+32): two D tiles.
// Grid.x = BATCH * PTILES * SPLIT.
// ---------------------------------------------------------------------------
__global__ __launch_bounds__(256) void splitpool_wmma(
    const float* __restrict__ x,
    const int* __restrict__ chunkStart,
    const int* __restrict__ ss,
    const int* __restrict__ npk,
    float* __restrict__ dest,
    int totalChunks, int use_atomic) {
  __shared__ int sbnd[17];            // row boundaries of the 16 chunks
  __shared__ unsigned short seg[448]; // local segment id per row of this split

  const int bid = blockIdx.x;
  const int s   = bid % SPLIT;
  const int tmp = bid / SPLIT;
  const int p   = tmp % PTILES;       // peak tile (16 output slots)
  const int i   = tmp / PTILES;       // sample
  const int t   = threadIdx.x;

  const int np = npk[i];
  const int c0 = ss[i] + p * 16;      // first global chunk of this tile
  int nvalid = np - p * 16;
  nvalid = nvalid < 0 ? 0 : (nvalid > 16 ? 16 : nvalid);

  if (t < 17) {
    int jj  = t < nvalid ? t : nvalid;            // clamp invalid slots -> empty
    int idx = c0 + jj;
    if (idx > totalChunks) idx = totalChunks;
    sbnd[t] = chunkStart[idx];
  }
  __syncthreads();

  const int rowStart  = sbnd[0];
  const int rowEnd    = sbnd[16];
  const int len       = rowEnd - rowStart;
  const int stepTotal = (len + 3) >> 2;
  const int stepsPer  = (stepTotal + SPLIT - 1) / SPLIT;
  const int s0        = s * stepsPer;
  const int s1raw     = s0 + stepsPer;
  const int s1        = s1raw < stepTotal ? s1raw : stepTotal;
  const int mySteps   = s1 > s0 ? s1 - s0 : 0;
  const int myRowStart = rowStart + 4 * s0;
  const int myRows     = 4 * mySteps;             // <= 436 < 448

  // Per-row local segment id (16 = no chunk in this tile -> A contributes 0)
  for (int idx = t; idx < myRows; idx += 256) {
    const int r = myRowStart + idx;
    int m = 0;
#pragma unroll
    for (int j = 1; j <= 16; ++j) m += (r >= sbnd[j]);
    seg[idx] = (unsigned short)m;
  }
  __syncthreads();

  const int wave  = t >> 5;
  const int lane  = t & 31;
  const int mloc  = lane & 15;   // chunk row this lane accumulates in A
  const int khalf = lane >> 4;   // 0: K=0,1   1: K=2,3  (ISA A/B VGPR layout)
  const int n     = lane & 15;   // column within tile for B
  const int colA  = wave * 32 + n;
  const int colB  = colA + 16;

  v8f accA = {0.f,0.f,0.f,0.f,0.f,0.f,0.f,0.f};
  v8f accB = {0.f,0.f,0.f,0.f,0.f,0.f,0.f,0.f};
  const int maxRow = BATCH * LEN - 1;

  for (int st = 0; st < mySteps; ++st) {
    const int lbase = 4 * st + khalf * 2;         // local row index for K pair
    const int ra = myRowStart + lbase;            // row for a.x / b.x
    const int rb = ra + 1;                        // row for a.y / b.y
    const int sa = seg[lbase];
    const int sc = seg[lbase + 1];
    v2f a;
    a.x = (sa == mloc) ? 1.0f : 0.0f;
    a.y = (sc == mloc) ? 1.0f : 0.0f;
    const long rac = (ra > maxRow) ? maxRow : ra; // tail rows: A==0 anyway
    const long rbc = (rb > maxRow) ? maxRow : rb;
    const float* pa = x + rac * DIM;
    const float* pb = x + rbc * DIM;
    v2f b0, b1;
    b0.x = __builtin_nontemporal_load(pa + colA);
    b0.y = __builtin_nontemporal_load(pb + colA);
    b1.x = __builtin_nontemporal_load(pa + colB);
    b1.y = __builtin_nontemporal_load(pb + colB);
    accA = __builtin_amdgcn_wmma_f32_16x16x4_f32(
        false, a, false, b0, (short)0, accA, false, false);
    accB = __builtin_amdgcn_wmma_f32_16x16x4_f32(
        false, a, false, b1, (short)0, accB, false, false);
  }

  // C/D layout: VGPR v -> M=v (lanes 0-15), M=v+8 (lanes 16-31); N=lane&15.
  if (use_atomic) {
#pragma unroll
    for (int v = 0; v < 8; ++v) {
      const int mrow = v + khalf * 8;
      if (mrow < nvalid) {
        const int j = p * 16 + mrow;
        const size_t o = ((size_t)i * PMAX + j) * DIM;
        atomicAdd(dest + o + colA, accA[v]);
        atomicAdd(dest + o + colB, accB[v]);
      }
    }
  } else {
    // deterministic path: write full tile (zeros included) to partials slice
#pragma unroll
    for (int v = 0; v < 8; ++v) {
      const int mrow = v + khalf * 8;
      const int j = p * 16 + mrow;
      const size_t o = (((size_t)s * BATCH + i) * PMAX + j) * DIM;
      dest[o + colA] = accA[v];
      dest[o + colB] = accB[v];
    }
  }
}

// ---------------------------------------------------------------------------
// Kernel 3: deterministic fixed-order reduction of SPLIT partials -> out.
// ---------------------------------------------------------------------------
__global__ __launch_bounds__(256) void reduce_kernel(
    const float* __restrict__ part, float* __restrict__ out) {
  const size_t e = (size_t)blockIdx.x * 256 + threadIdx.x;
  float v = 0.0f;
#pragma unroll
  for (int s = 0; s < SPLIT; ++s) v += part[(size_t)s * OUT_N + e];
  out[e] = v;
}

// ---------------------------------------------------------------------------
extern "C" void kernel_launch(void* const* d_in, const int* in_sizes, int n_in,
                              void* d_out, int out_size, void* d_ws, size_t ws_size,
                              hipStream_t stream) {
  const float* x  = (const float*)d_in[0];
  const int*   cs = (const int*)d_in[1];   // chunk_size (int32)
  const int*   np = (const int*)d_in[2];   // n_peaks (int32)
  float* out = (float*)d_out;
  const int totalChunks = in_sizes[1];

  int* chunkStart = (int*)d_ws;                        // totalChunks+1 ints
  int* ssidx      = chunkStart + (totalChunks + 1);    // BATCH+1 ints
  size_t prefixBytes = (size_t)(totalChunks + 1 + BATCH + 1) * sizeof(int);
  size_t partOff     = (prefixBytes + 255) & ~(size_t)255;
  float* partial     = (float*)((char*)d_ws + partOff);
  size_t need        = partOff + (size_t)SPLIT * OUT_N * sizeof(float);

  prefix_kernel<<<1, 256, 0, stream>>>(cs, np, chunkStart, ssidx, totalChunks);

  dim3 grid(BATCH * PTILES * SPLIT);
  if (ws_size >= need) {
    splitpool_wmma<<<grid, 256, 0, stream>>>(x, chunkStart, ssidx, np,
                                             partial, totalChunks, 0);
    reduce_kernel<<<OUT_N / 256, 256, 0, stream>>>(partial, out);
  } else {
    zero_kernel<<<OUT_N / 256, 256, 0, stream>>>(out);
    splitpool_wmma<<<grid, 256, 0, stream>>>(x, chunkStart, ssidx, np,
                                             out, totalChunks, 1);
  }
}